// BearingQCCFeatureMotion_76836964926187
// MI455X (gfx1250) — compile-verified
//
#include <hip/hip_runtime.h>
#include <math.h>

// ---------------- problem constants (from reference setup) ----------------
constexpr int B = 2, F = 8, P = 2048, T = F - 1;
constexpr int K40 = 40, K15 = 15, K5 = 5;

// ---------------- workspace layout (floats) ----------------
constexpr size_t OFF_CENT   = 0;                           // B*F*4       = 64
constexpr size_t OFF_BQ     = OFF_CENT + (size_t)B*F*4;    // B*F*P*4     = 131072
constexpr size_t OFF_QF     = OFF_BQ   + (size_t)B*F*P*4;  // B*T*P*4     = 114688
constexpr size_t OFF_G5     = OFF_QF   + (size_t)B*T*P*4;  // B*T*P
constexpr size_t OFF_G15    = OFF_G5   + (size_t)B*T*P;
constexpr size_t OFF_G40    = OFF_G15  + (size_t)B*T*P;
constexpr size_t OFF_MINC   = OFF_G40  + (size_t)B*T*P;    // 3*B*P
constexpr size_t OFF_PARAMS = OFF_MINC + (size_t)3*B*P;    // 6

typedef float v2f __attribute__((ext_vector_type(2)));
typedef float v8f __attribute__((ext_vector_type(8)));

// ============================================================
// Kernel 1: per-(b,f) bbox centroid (min/max over P points)
// ============================================================
__global__ __launch_bounds__(256) void qcc_centroid_kernel(
    const float* __restrict__ pts4d, float* __restrict__ cent)
{
    __shared__ float smin[256][3];
    __shared__ float smax[256][3];
    const int bf  = blockIdx.x;          // b*F + f
    const int tid = threadIdx.x;
    const float4* src = (const float4*)(pts4d) + (size_t)bf * P;

    float mn[3] = { 3.4e38f,  3.4e38f,  3.4e38f};
    float mx[3] = {-3.4e38f, -3.4e38f, -3.4e38f};
    for (int i = tid; i < P; i += 256) {
        float4 v = src[i];
        mn[0] = fminf(mn[0], v.x); mx[0] = fmaxf(mx[0], v.x);
        mn[1] = fminf(mn[1], v.y); mx[1] = fmaxf(mx[1], v.y);
        mn[2] = fminf(mn[2], v.z); mx[2] = fmaxf(mx[2], v.z);
    }
    for (int c = 0; c < 3; ++c) { smin[tid][c] = mn[c]; smax[tid][c] = mx[c]; }
    __syncthreads();
    for (int st = 128; st > 0; st >>= 1) {
        if (tid < st) {
            for (int c = 0; c < 3; ++c) {
                smin[tid][c] = fminf(smin[tid][c], smin[tid + st][c]);
                smax[tid][c] = fmaxf(smax[tid][c], smax[tid + st][c]);
            }
        }
        __syncthreads();
    }
    if (tid == 0) {
        for (int c = 0; c < 3; ++c)
            cent[bf * 4 + c] = 0.5f * (smin[0][c] + smax[0][c]);
    }
}

// ============================================================
// Kernel 2: bearing quaternions bq[b][f][p] (w,x,y,z)
// ============================================================
__global__ __launch_bounds__(256) void qcc_bq_kernel(
    const float* __restrict__ pts4d, const float* __restrict__ cent,
    float* __restrict__ bq)
{
    int i = blockIdx.x * blockDim.x + threadIdx.x;
    if (i >= B * F * P) return;
    const int p  = i % P;
    const int bf = i / P;
    float4 v = ((const float4*)pts4d)[(size_t)bf * P + p];
    float dx = v.x - cent[bf * 4 + 0];
    float dy = v.y - cent[bf * 4 + 1];
    float dz = v.z - cent[bf * 4 + 2];
    float n = fmaxf(sqrtf(dx * dx + dy * dy + dz * dz), 1e-12f);
    dx /= n; dy /= n; dz /= n;
    float dot  = fminf(fmaxf(dy, -1.0f + 1e-7f), 1.0f - 1e-7f);
    float half = 0.5f * acosf(dot);
    float ax = dz, az = -dx;                      // axis = [dz, 0, -dx]
    float an = fmaxf(sqrtf(ax * ax + az * az), 1e-12f);
    float s  = sinf(half);
    float* o = bq + (size_t)i * 4;
    o[0] = cosf(half);
    o[1] = (ax / an) * s;
    o[2] = 0.0f;
    o[3] = (az / an) * s;
}

// ============================================================
// Kernel 3: forward relative quaternions q_fwd[b][t][p]
// ============================================================
__global__ __launch_bounds__(256) void qcc_qfwd_kernel(
    const float* __restrict__ bq, float* __restrict__ qf)
{
    int i = blockIdx.x * blockDim.x + threadIdx.x;
    if (i >= B * T * P) return;
    const int p  = i % P;
    const int bt = i / P;
    const int b  = bt / T;
    const int t  = bt % T;
    const float* qs = bq + ((size_t)(b * F + t)     * P + p) * 4; // frame t
    const float* qt = bq + ((size_t)(b * F + t + 1) * P + p) * 4; // frame t+1
    // q_src = conj(bq[t]); q_fwd = hamilton(q_tgt, q_src)
    float bw = qs[0], bx = -qs[1], by = -qs[2], bz = -qs[3];
    float aw = qt[0], ax =  qt[1], ay =  qt[2], az =  qt[3];
    float w = aw * bw - ax * bx - ay * by - az * bz;
    float x = aw * bx + ax * bw + ay * bz - az * by;
    float y = aw * by - ax * bz + ay * bw + az * bx;
    float z = aw * bz + ax * by - ay * bx + az * bw;
    float n = fmaxf(sqrtf(w * w + x * x + y * y + z * z), 1e-12f);
    float* o = qf + (size_t)i * 4;
    o[0] = w / n; o[1] = x / n; o[2] = y / n; o[3] = z / n;
}

// ============================================================
// Kernel 4: fused WMMA distance field + split top-40 + geo sums
//   - one wave32 per 16-row block; 2048-point set staged in LDS
//   - 4 back-to-back v_wmma_f32_16x16x4_f32 per barrier (64 cols)
//   - branchless fragment loads: every lane does one ds_load_b64
//     (lo lanes read (x,y), hi lanes read (z,w)), then cndmask
//     zeroes the K=3 pad component for hi lanes
//   - all 32 lanes select: lane r scans cols [0,32) of the strip,
//     lane r+16 scans [32,64); two sorted top-40 lists per row are
//     merged on the fly (stable, matches jax.lax.top_k ordering)
// ============================================================
__global__ __launch_bounds__(32) void qcc_dist_topk_kernel(
    const float* __restrict__ pts4d, const float* __restrict__ qf,
    float* __restrict__ g5, float* __restrict__ g15, float* __restrict__ g40)
{
    constexpr int UN = 4;                 // column tiles per barrier
    __shared__ float4 pl[P];              // x, y, z, |p|^2  (32 KB)
    __shared__ float  dtile[16][UN*16+1]; // 16 x 64 neg_dist strip (padded)
    __shared__ float  topv[32][K40];      // per (row, half) sorted values desc
    __shared__ int    topi[32][K40];      // matching column indices

    const int blk = blockIdx.x;           // (b*T + t) * 128 + rowblock
    const int rb  = blk & 127;
    const int bt  = blk >> 7;             // b*T + t
    const int b   = bt / T;
    const int t   = bt % T;
    const int tid = threadIdx.x;

    // Stage this (b,t)'s points into LDS once (reused by all column strips).
    const float4* src = (const float4*)(pts4d) + (size_t)(b * F + t) * P;
    for (int i = tid; i < P; i += 32) {
        float4 v = src[i];
        v.w = v.x * v.x + v.y * v.y + v.z * v.z;   // squared norm
        pl[i] = v;
    }
    for (int j = 0; j < K40; ++j) { topv[tid][j] = -3.4e38f; topi[tid][j] = 0; }
    // Warm L2/WGP$ with the quaternion table used in the gather phase.
    __builtin_prefetch(qf + ((size_t)bt * P + (size_t)tid * 64) * 4, 0, 1);
    __syncthreads();

    const int rbase  = rb * 16;
    const int lane15 = tid & 15;
    const int hi     = tid >> 4;          // 0: K=0/1 half, 1: K=2/3 half
    const float2* plh = (const float2*)pl;  // float4 viewed as two float2 halves

    // A fragment (16x4 f32): lanes 0-15 hold (x,y) [K=0,1],
    // lanes 16-31 hold (z, 0) [K=2, K=3 zero pad]. One ds_load_b64 per lane.
    v2f afrag;
    {
        float2 av = plh[(size_t)(rbase + lane15) * 2 + hi];
        afrag.x = av.x;
        afrag.y = hi ? 0.0f : av.y;
    }

    const int Mbase = hi * 8;
    const int N     = lane15;
    // Loop-invariant row squared norms for this lane's 8 output rows.
    float sqrow[8];
#pragma unroll
    for (int v = 0; v < 8; ++v) sqrow[v] = pl[rbase + Mbase + v].w;

    // Selection state: lane owns (row = tid&15, half = tid>>4).
    const int row   = lane15;
    const int chalf = hi * (UN * 16 / 2);  // 0 or 32 within strip
    float* tv = topv[tid];
    int*   ti = topi[tid];
    float  thr = -3.4e38f;                 // register-cached 40th value

    for (int ct = 0; ct < P / (UN * 16); ++ct) {
        const int cbase = ct * (UN * 16);
#pragma unroll
        for (int u = 0; u < UN; ++u) {
            const int cb = cbase + u * 16;
            // B fragment (4x16 f32): branchless single ds_load_b64 per lane.
            v2f bfrag;
            {
                float2 bv = plh[(size_t)(cb + lane15) * 2 + hi];
                bfrag.x = bv.x;
                bfrag.y = hi ? 0.0f : bv.y;
            }

            v8f c = {};
            c = __builtin_amdgcn_wmma_f32_16x16x4_f32(
                    /*neg_a=*/false, afrag, /*neg_b=*/false, bfrag,
                    /*c_mod=*/(short)0, c, /*reuse_a=*/false, /*reuse_b=*/false);

            // neg_dist[M][N] = 2*dot - |row|^2 - |col|^2
            const float sqc = pl[cb + N].w;
#pragma unroll
            for (int v = 0; v < 8; ++v)
                dtile[Mbase + v][u * 16 + N] = 2.0f * c[v] - sqrow[v] - sqc;
        }
        __syncthreads();

        // Sorted top-40 insertion over this lane's 32 columns
        // (stable: ties keep lower column index first).
#pragma unroll 1
        for (int j = 0; j < UN * 16 / 2; ++j) {
            const float val = dtile[row][chalf + j];
            if (val > thr) {
                int pos = K40 - 1;
                while (pos > 0 && tv[pos - 1] < val) {
                    tv[pos] = tv[pos - 1];
                    ti[pos] = ti[pos - 1];
                    --pos;
                }
                tv[pos] = val;
                ti[pos] = cbase + chalf + j;
                thr = tv[K40 - 1];
            }
        }
        __syncthreads();
    }

    // Merge the two sorted half-lists per row on the fly (value desc,
    // index asc on ties == jax.lax.top_k stable order) while computing
    // geodesic-angle prefix sums for k = 5 / 15 / 40.
    if (tid < 16) {
        const int r = rbase + tid;
        const float4* qfv = (const float4*)qf;
        const float4 qv = qfv[(size_t)bt * P + r];
        const float qw = qv.x, qx = qv.y, qy = qv.z, qz = qv.w;
        const float* va = topv[tid];       const int* na = topi[tid];
        const float* vb = topv[tid + 16];  const int* nb = topi[tid + 16];
        int ia = 0, ib = 0;
        float s5 = 0.0f, s15 = 0.0f, s40 = 0.0f;
#pragma unroll 1
        for (int j = 0; j < K40; ++j) {
            const float A  = va[ia];
            const float Bv = vb[ib];
            const bool takeA = (A > Bv) || ((A == Bv) && (na[ia] <= nb[ib]));
            const int idx = takeA ? na[ia] : nb[ib];
            ia += takeA ? 1 : 0;
            ib += takeA ? 0 : 1;
            const float4 nq = qfv[(size_t)bt * P + idx];
            float d = fabsf(qw * nq.x + qx * nq.y + qy * nq.z + qz * nq.w);
            d = fminf(fmaxf(d, 0.0f), 1.0f - 1e-7f);
            const float g = 2.0f * acosf(d);
            s40 += g;
            if (j < K15) s15 += g;
            if (j < K5)  s5  += g;
        }
        const size_t o = (size_t)bt * P + r;
        g5[o] = s5; g15[o] = s15; g40[o] = s40;
    }
}

// ============================================================
// Kernel 5: mean over t and over k -> mean_inc[scale][b][p]
// ============================================================
__global__ __launch_bounds__(256) void qcc_meaninc_kernel(
    const float* __restrict__ g5, const float* __restrict__ g15,
    const float* __restrict__ g40, float* __restrict__ minc)
{
    int i = blockIdx.x * blockDim.x + threadIdx.x;
    if (i >= 3 * B * P) return;
    const int s  = i / (B * P);
    const int bp = i % (B * P);
    const int b  = bp / P, p = bp % P;
    const float* g = (s == 0) ? g5 : (s == 1) ? g15 : g40;
    const float kk = (s == 0) ? 5.0f : (s == 1) ? 15.0f : 40.0f;
    float acc = 0.0f;
    for (int t = 0; t < T; ++t) acc += g[(size_t)(b * T + t) * P + p];
    minc[(size_t)s * (B * P) + bp] = acc / (kk * (float)T);
}

// ============================================================
// Kernel 6: per-scale lower-median + max via in-LDS bitonic sort
// ============================================================
__global__ __launch_bounds__(1024) void qcc_median_kernel(
    const float* __restrict__ minc, float* __restrict__ params)
{
    __shared__ float sd[B * P];          // 4096 floats = 16 KB
    const int s   = blockIdx.x;
    const int tid = threadIdx.x;
    const int N   = B * P;
    const float* src = minc + (size_t)s * N;
    for (int i = tid; i < N; i += 1024) sd[i] = src[i];
    __syncthreads();
    for (int k = 2; k <= N; k <<= 1) {
        for (int j = k >> 1; j > 0; j >>= 1) {
            for (int i = tid; i < N; i += 1024) {
                const int ixj = i ^ j;
                if (ixj > i) {
                    const float a = sd[i], c = sd[ixj];
                    const bool up = ((i & k) == 0);
                    if ((a > c) == up) { sd[i] = c; sd[ixj] = a; }
                }
            }
            __syncthreads();
        }
    }
    if (tid == 0) {
        params[2 * s + 0] = fmaxf(sd[(N - 1) / 2], 1e-6f);  // scale
        params[2 * s + 1] = sd[N - 1];                      // global max
    }
}

// ============================================================
// Kernel 7: rigidities -> output (B, 3, F*P)
// ============================================================
__global__ __launch_bounds__(256) void qcc_out_kernel(
    const float* __restrict__ minc, const float* __restrict__ params,
    float* __restrict__ out, int n)
{
    int i = blockIdx.x * blockDim.x + threadIdx.x;
    if (i >= n) return;
    const int p = i % P;
    const int s = (i / (P * F)) % 3;
    const int b = i / (P * F * 3);
    const float m     = minc[(size_t)s * (B * P) + b * P + p];
    const float scale = params[2 * s + 0];
    const float mx    = params[2 * s + 1];
    out[i] = (mx > 0.0f) ? expf(-m / scale) : 1.0f;
}

// ============================================================
extern "C" void kernel_launch(void* const* d_in, const int* in_sizes, int n_in,
                              void* d_out, int out_size, void* d_ws, size_t ws_size,
                              hipStream_t stream)
{
    const float* pts4d = (const float*)d_in[0];
    float* ws     = (float*)d_ws;
    float* cent   = ws + OFF_CENT;
    float* bq     = ws + OFF_BQ;
    float* qf     = ws + OFF_QF;
    float* g5     = ws + OFF_G5;
    float* g15    = ws + OFF_G15;
    float* g40    = ws + OFF_G40;
    float* minc   = ws + OFF_MINC;
    float* params = ws + OFF_PARAMS;

    qcc_centroid_kernel<<<B * F, 256, 0, stream>>>(pts4d, cent);
    qcc_bq_kernel<<<(B * F * P + 255) / 256, 256, 0, stream>>>(pts4d, cent, bq);
    qcc_qfwd_kernel<<<(B * T * P + 255) / 256, 256, 0, stream>>>(bq, qf);
    qcc_dist_topk_kernel<<<B * T * (P / 16), 32, 0, stream>>>(pts4d, qf, g5, g15, g40);
    qcc_meaninc_kernel<<<(3 * B * P + 255) / 256, 256, 0, stream>>>(g5, g15, g40, minc);
    qcc_median_kernel<<<3, 1024, 0, stream>>>(minc, params);
    qcc_out_kernel<<<(B * 3 * F * P + 255) / 256, 256, 0, stream>>>(
        minc, params, (float*)d_out, B * 3 * F * P);
}